// MMGCN_89867895701861
// MI455X (gfx1250) — compile-verified
//
#include <hip/hip_runtime.h>
#include <math.h>

// ---------------------------------------------------------------------------
// MMGCN forward for MI455X (gfx1250, wave32).
//
// Input order assumption (setup_inputs() insertion order; nested params dict
// flattened in its own insertion order):
//   0 x_m(215*128) 1 x_d(110*128) 2 mi_gua 3 mi_cos 4 mi_fun
//   5 di_gua 6 di_cos 7 di_sem
//   8..10 mi_*_edges(2*EM) 11..13 di_*_edges(2*ED) 14 train_edges(P*2)
//   15 Wx1f 16 bx1f 17 Wx2f 18 bx2f 19 Wx1s 20 bx1s 21 Wx2s 22 bx2s
//   23 Wy1f 24 by1f 25 Wy2f 26 by2f 27 Wy1s 28 by1s 29 Wy2s 30 by2s
//   31 fc1x_w 32 fc1x_b 33 fc2x_w 34 fc2x_b 35 fc1y_w 36 fc1y_b
//   37 fc2y_w 38 fc2y_b 39 cnnx_w 40 cnnx_b 41 cnny_w 42 cnny_b
//   43 lin_w 44 lin_b
// ---------------------------------------------------------------------------

typedef float v2f __attribute__((ext_vector_type(2)));
typedef float v8f __attribute__((ext_vector_type(8)));

#define NM 215
#define ND 110
#define FF 128
#define MPX 224   // NM padded to multiple of 16 (and of 32 for K-chunking)
#define MPY 112   // ND padded

static inline int cdiv(int a, int b) { return (a + b - 1) / b; }

// ---------------------------------------------------------------------------
__global__ void mmgcn_fill(float* __restrict__ p, float v, int n) {
  int i = blockIdx.x * blockDim.x + threadIdx.x;
  if (i < n) p[i] = v;
}

// copy Nn*FF input rows into an Mpad*FF buffer, zeroing the pad rows
__global__ void mmgcn_copy_pad(const float* __restrict__ in,
                               float* __restrict__ out, int Nn, int Mpad) {
  int i = blockIdx.x * blockDim.x + threadIdx.x;
  if (i >= Mpad * FF) return;
  int row = i >> 7;
  out[i] = (row < Nn) ? in[i] : 0.0f;
}

// Dense adjacency A[d*lda+s] += sim[s*Nn+d] over the edge list.
// Deterministic despite atomics: all contributions to one cell are the same
// value sim[s,d], so summation order cannot change the rounded result.
__global__ void mmgcn_scatter_adj(const int* __restrict__ ei,
                                  const float* __restrict__ sim,
                                  float* __restrict__ A, int E, int Nn,
                                  int lda) {
  int e = blockIdx.x * blockDim.x + threadIdx.x;
  if (e >= E) return;
  int s = ei[e];
  int d = ei[E + e];
  atomicAdd(&A[d * lda + s], sim[s * Nn + d]);
}

// deg[d] = 1 (self loop) + row-sum(A);  dinv = deg^-1/2
__global__ void mmgcn_row_dinv(const float* __restrict__ A,
                               float* __restrict__ dinv, int Nn, int lda) {
  int r = blockIdx.x * blockDim.x + threadIdx.x;
  if (r >= Nn) return;
  float s = 1.0f;
  for (int c = 0; c < Nn; ++c) s += A[r * lda + c];
  dinv[r] = (s > 0.0f) ? rsqrtf(s) : 0.0f;
}

// In-place: A[d,s] = dinv[d]*A[d,s]*dinv[s]  (+ dinv[d]^2 on diagonal).
// Pad region of A stays zero, so the padded GEMM needs no bounds checks.
__global__ void mmgcn_norm_adj(float* __restrict__ A,
                               const float* __restrict__ dinv, int Nn,
                               int lda) {
  int i = blockIdx.x * blockDim.x + threadIdx.x;
  if (i >= Nn * Nn) return;
  int d = i / Nn;
  int s = i - d * Nn;
  float v = dinv[d] * A[d * lda + s] * dinv[s];
  if (d == s) v += dinv[d] * dinv[d];
  A[d * lda + s] = v;
}

// ---------------------------------------------------------------------------
// f32 WMMA GEMM, fully padded: C[Mpad,128] = A[Mpad,K] @ B[K,128].
// lda is A's row stride; ldb = ldc = 128. K must be a multiple of 32.
// Block = 256 threads = 8 waves; block computes one 16x128 stripe
// (wave w owns columns [16w,16w+16)). K consumed in 32-wide chunks staged
// through LDS; the inner loop is branch-free ds_load + V_WMMA_F32_16X16X4_F32.
//
// Operand layouts per CDNA5 ISA (CDNA5 ISA):
//   A 16x4 : lanes 0-15 M=lane, VGPR0/1 = K 0/1; lanes 16-31 K 2/3.
//   B 4x16 : N = lane&15; VGPR0/1 = K 0/1 (lanes<16), K 2/3 (lanes>=16).
//   C 16x16: VGPR v -> row v (+8 for lanes>=16), col = lane&15.
#define KC 32
#define SA_STR 34    // 16x32 A tile, padded row stride (bank spread)
#define SB_STR 132   // 32x128 B tile, padded row stride

__global__ __launch_bounds__(256) void mmgcn_wmma_gemm(
    const float* __restrict__ A, const float* __restrict__ B,
    float* __restrict__ C, int K, int lda) {
  __shared__ float sA[16 * SA_STR];
  __shared__ float sB[KC * SB_STR];

  const int t    = threadIdx.x;
  const int wave = t >> 5;
  const int lane = t & 31;
  const int half = lane >> 4;
  const int l16  = lane & 15;
  const int row0 = blockIdx.x * 16;
  const int col  = wave * 16 + l16;   // absolute output column (N = 128)

  v8f acc = {0.f, 0.f, 0.f, 0.f, 0.f, 0.f, 0.f, 0.f};

  // staging indices (constant across chunks)
  const int ar = t >> 4;              // A tile row      0..15
  const int ac = (t & 15) * 2;        // A tile col pair 0..30

  for (int k0 = 0; k0 < K; k0 += KC) {
    // ---- stage A 16xKC: one v2f per thread ----
    v2f av = *(const v2f*)(A + (row0 + ar) * lda + k0 + ac);
    *(v2f*)(sA + ar * SA_STR + ac) = av;

    // ---- stage B KCx128: four float4 per thread ----
#pragma unroll
    for (int i = 0; i < 4; ++i) {
      int idx = t + i * 256;          // 0..1023 float4 slots
      int kr  = idx >> 5;             // 0..31
      int c4  = (idx & 31) * 4;       // 0..124
      float4 bv = *(const float4*)(B + (k0 + kr) * FF + c4);
      *(float4*)(sB + kr * SB_STR + c4) = bv;
    }

    // prefetch next B chunk (global_prefetch_b8), one 64B line per thread
    if (k0 + KC < K) {
      const char* nb = (const char*)(B + (k0 + KC) * FF);
      __builtin_prefetch(nb + t * 64, 0, 0);
    }

    __syncthreads();

    // ---- 8 branch-free WMMA steps over this chunk ----
#pragma unroll
    for (int kk = 0; kk < KC; kk += 4) {
      int kb = kk + half * 2;
      v2f a = *(const v2f*)(sA + l16 * SA_STR + kb);
      v2f b;
      b.x = sB[(kb    ) * SB_STR + col];
      b.y = sB[(kb + 1) * SB_STR + col];
      acc = __builtin_amdgcn_wmma_f32_16x16x4_f32(
          false, a, false, b, (short)0, acc, false, false);
    }
    __syncthreads();
  }

  // ---- store 16x16 tile (all rows exist: C has Mpad rows) ----
  const int rbase = row0 + half * 8;
#pragma unroll
  for (int v = 0; v < 8; ++v) C[(rbase + v) * FF + col] = acc[v];
}

// out = relu(agg + bias[f]) over the whole padded buffer (pad rows get
// relu(bias); they are multiplied by Bhat's zero pad columns downstream)
__global__ void mmgcn_bias_relu(const float* __restrict__ agg,
                                const float* __restrict__ bias,
                                float* __restrict__ out, int n) {
  int i = blockIdx.x * blockDim.x + threadIdx.x;
  if (i >= n) return;
  out[i] = fmaxf(agg[i] + bias[i & (FF - 1)], 0.0f);
}

// ---------------------------------------------------------------------------
// channel attention: pooled[c] = mean over the Nn*FF valid elements
__global__ void mmgcn_pool_mean(const float* __restrict__ feats,
                                float* __restrict__ pooled, int NFel,
                                int chanStride) {
  __shared__ float red[256];
  int c = blockIdx.x;
  const float* f = feats + (size_t)c * chanStride;
  float s = 0.0f;
  for (int i = threadIdx.x; i < NFel; i += blockDim.x) s += f[i];
  red[threadIdx.x] = s;
  __syncthreads();
  for (int w = 128; w > 0; w >>= 1) {
    if (threadIdx.x < w) red[threadIdx.x] += red[threadIdx.x + w];
    __syncthreads();
  }
  if (threadIdx.x == 0) pooled[c] = red[0] / (float)NFel;
}

__global__ void mmgcn_attn_mlp(const float* __restrict__ pooled,
                               const float* __restrict__ fc1w,
                               const float* __restrict__ fc1b,
                               const float* __restrict__ fc2w,
                               const float* __restrict__ fc2b,
                               float* __restrict__ svec) {
  __shared__ float a[30];
  int t = threadIdx.x;
  if (t < 30) {
    float s = fc1b[t];
    for (int c = 0; c < 6; ++c) s += pooled[c] * fc1w[t * 6 + c];
    a[t] = fmaxf(s, 0.0f);
  }
  __syncthreads();
  if (t < 6) {
    float s = fc2b[t];
    for (int j = 0; j < 30; ++j) s += a[j] * fc2w[t * 30 + j];
    svec[t] = 1.0f / (1.0f + expf(-s));
  }
}

// out[n,f] = sum_c cnnw[c]*relu(s[c]*feat_c[n,f]) + cnnb; optionally *lin_w[f]
__global__ void mmgcn_combine(const float* __restrict__ feats,
                              const float* __restrict__ svec,
                              const float* __restrict__ cnnw,
                              const float* __restrict__ cnnb,
                              const float* __restrict__ linw,  // may be null
                              float* __restrict__ out, int NFel,
                              int chanStride) {
  int i = blockIdx.x * blockDim.x + threadIdx.x;
  if (i >= NFel) return;
  float acc = cnnb[0];
#pragma unroll
  for (int c = 0; c < 6; ++c)
    acc += cnnw[c] * fmaxf(svec[c] * feats[(size_t)c * chanStride + i], 0.0f);
  if (linw) acc *= linw[i & (FF - 1)];
  out[i] = acc;
}

// ---------------------------------------------------------------------------
// Pair scoring: both feature tables staged in LDS (~168 KB < 320 KB/WGP).
// Row stride 132 (mod 64 == 4) spreads random-row reads across banks while
// keeping float4 rows 16B-aligned.
#define SX_STRIDE 132
#define SMEM_FLOATS ((NM + ND) * SX_STRIDE)

__global__ void mmgcn_pair_score(const int* __restrict__ pairs,
                                 const float* __restrict__ zx,  // x*lin_w
                                 const float* __restrict__ yf,
                                 const float* __restrict__ lin_b,
                                 float* __restrict__ out, int P) {
  extern __shared__ float smem[];
  float* sx = smem;
  float* sy = smem + NM * SX_STRIDE;
  for (int i = threadIdx.x; i < NM * FF; i += blockDim.x) {
    int r = i >> 7, f = i & 127;
    sx[r * SX_STRIDE + f] = zx[i];
  }
  for (int i = threadIdx.x; i < ND * FF; i += blockDim.x) {
    int r = i >> 7, f = i & 127;
    sy[r * SX_STRIDE + f] = yf[i];
  }
  __syncthreads();
  float bias = lin_b[0];
  int tid = blockIdx.x * blockDim.x + threadIdx.x;
  int stride = gridDim.x * blockDim.x;
  for (int p = tid; p < P; p += stride) {
    int m = pairs[2 * p];
    int d = pairs[2 * p + 1];
    const float* xr = &sx[m * SX_STRIDE];
    const float* yr = &sy[d * SX_STRIDE];
    float acc = 0.0f;
#pragma unroll
    for (int f = 0; f < FF; f += 4) {
      float4 xv = *(const float4*)(xr + f);
      float4 yv = *(const float4*)(yr + f);
      acc += xv.x * yv.x + xv.y * yv.y + xv.z * yv.z + xv.w * yv.w;
    }
    float t = acc + bias;
    out[p] = 1.0f / (1.0f + expf(-t));
  }
}

// ---------------------------------------------------------------------------
extern "C" void kernel_launch(void* const* d_in, const int* in_sizes, int n_in,
                              void* d_out, int out_size, void* d_ws,
                              size_t ws_size, hipStream_t stream) {
  if (n_in < 45) return;

  const float* x_m = (const float*)d_in[0];
  const float* x_d = (const float*)d_in[1];
  const float* mi_sim[3] = {(const float*)d_in[2], (const float*)d_in[3],
                            (const float*)d_in[4]};
  const float* di_sim[3] = {(const float*)d_in[5], (const float*)d_in[6],
                            (const float*)d_in[7]};
  const int* mi_e[3] = {(const int*)d_in[8], (const int*)d_in[9],
                        (const int*)d_in[10]};
  const int* di_e[3] = {(const int*)d_in[11], (const int*)d_in[12],
                        (const int*)d_in[13]};
  const int* train = (const int*)d_in[14];
  const int EM = in_sizes[8] / 2;
  const int ED = in_sizes[11] / 2;
  const int P  = in_sizes[14] / 2;

  const float* Wx1f = (const float*)d_in[15]; const float* bx1f = (const float*)d_in[16];
  const float* Wx2f = (const float*)d_in[17]; const float* bx2f = (const float*)d_in[18];
  const float* Wx1s = (const float*)d_in[19]; const float* bx1s = (const float*)d_in[20];
  const float* Wx2s = (const float*)d_in[21]; const float* bx2s = (const float*)d_in[22];
  const float* Wy1f = (const float*)d_in[23]; const float* by1f = (const float*)d_in[24];
  const float* Wy2f = (const float*)d_in[25]; const float* by2f = (const float*)d_in[26];
  const float* Wy1s = (const float*)d_in[27]; const float* by1s = (const float*)d_in[28];
  const float* Wy2s = (const float*)d_in[29]; const float* by2s = (const float*)d_in[30];
  const float* fc1x_w = (const float*)d_in[31]; const float* fc1x_b = (const float*)d_in[32];
  const float* fc2x_w = (const float*)d_in[33]; const float* fc2x_b = (const float*)d_in[34];
  const float* fc1y_w = (const float*)d_in[35]; const float* fc1y_b = (const float*)d_in[36];
  const float* fc2y_w = (const float*)d_in[37]; const float* fc2y_b = (const float*)d_in[38];
  const float* cnnx_w = (const float*)d_in[39]; const float* cnnx_b = (const float*)d_in[40];
  const float* cnny_w = (const float*)d_in[41]; const float* cnny_b = (const float*)d_in[42];
  const float* lin_w  = (const float*)d_in[43]; const float* lin_b  = (const float*)d_in[44];

  // workspace layout (floats); total ~1.8 MB — stays L2-resident
  float* ws = (float*)d_ws;
  const int NFX  = NM * FF;            // valid elements, x side
  const int NFY  = ND * FF;
  const int CSX  = MPX * FF;           // padded channel stride, x side
  const int CSY  = MPY * FF;
  size_t off = 0;
  float* featX = ws + off;  off += (size_t)6 * CSX;
  float* featY = ws + off;  off += (size_t)6 * CSY;
  float* xpadX = ws + off;  off += CSX;
  float* xpadY = ws + off;  off += CSY;
  float* hbuf  = ws + off;  off += CSX;          // max(CSX, CSY)
  float* aggb  = ws + off;  off += CSX;
  float* Abuf  = ws + off;  off += (size_t)MPX * MPX;
  float* dinv  = ws + off;  off += 256;
  float* zx    = ws + off;  off += NFX;
  float* yfin  = ws + off;  off += NFY;
  float* pooled = ws + off; off += 8;
  float* svec   = ws + off; off += 8;

  // one GCN layer: dense normalized adjacency + two padded WMMA GEMMs.
  // Mpad = padded node count (multiple of 16); Abuf uses lda = Mpad and its
  // pad region is zero, so no bounds checks exist anywhere in the GEMMs.
  auto gc = [&](const float* xin, const int* ei, const float* sim,
                const float* W, const float* bias, float* outF, int Nn, int E,
                int Mpad) {
    mmgcn_fill<<<cdiv(Mpad * Mpad, 256), 256, 0, stream>>>(Abuf, 0.0f,
                                                           Mpad * Mpad);
    mmgcn_scatter_adj<<<cdiv(E, 256), 256, 0, stream>>>(ei, sim, Abuf, E, Nn,
                                                        Mpad);
    mmgcn_row_dinv<<<cdiv(Nn, 128), 128, 0, stream>>>(Abuf, dinv, Nn, Mpad);
    mmgcn_norm_adj<<<cdiv(Nn * Nn, 256), 256, 0, stream>>>(Abuf, dinv, Nn,
                                                           Mpad);
    // h = x @ W            (Mpad x 128 x 128)
    mmgcn_wmma_gemm<<<Mpad / 16, 256, 0, stream>>>(xin, W, hbuf, FF, FF);
    // agg = Bhat @ h       (Mpad x 128 x Mpad); diag carries self-loop
    mmgcn_wmma_gemm<<<Mpad / 16, 256, 0, stream>>>(Abuf, hbuf, aggb, Mpad,
                                                   Mpad);
    mmgcn_bias_relu<<<cdiv(Mpad * FF, 256), 256, 0, stream>>>(aggb, bias, outF,
                                                              Mpad * FF);
  };

  // layer-0 inputs, zero-padded (never read outside harness allocations)
  mmgcn_copy_pad<<<cdiv(CSX, 256), 256, 0, stream>>>(x_m, xpadX, NM, MPX);
  mmgcn_copy_pad<<<cdiv(CSY, 256), 256, 0, stream>>>(x_d, xpadY, ND, MPY);

  float* fX[6]; for (int k = 0; k < 6; ++k) fX[k] = featX + (size_t)k * CSX;
  float* fY[6]; for (int k = 0; k < 6; ++k) fY[k] = featY + (size_t)k * CSY;

  // miRNA branch (g2 takes s1 as input, matching the reference quirk)
  gc(xpadX, mi_e[0], mi_sim[0], Wx1f, bx1f, fX[0], NM, EM, MPX);  // f1
  gc(fX[0], mi_e[0], mi_sim[0], Wx2f, bx2f, fX[1], NM, EM, MPX);  // f2
  gc(xpadX, mi_e[1], mi_sim[1], Wx1s, bx1s, fX[2], NM, EM, MPX);  // s1
  gc(fX[2], mi_e[1], mi_sim[1], Wx2s, bx2s, fX[3], NM, EM, MPX);  // s2
  gc(xpadX, mi_e[2], mi_sim[2], Wx1s, bx1s, fX[4], NM, EM, MPX);  // g1
  gc(fX[2], mi_e[2], mi_sim[2], Wx2s, bx2s, fX[5], NM, EM, MPX);  // g2 <- s1

  // disease branch
  gc(xpadY, di_e[0], di_sim[0], Wy1f, by1f, fY[0], ND, ED, MPY);
  gc(fY[0], di_e[0], di_sim[0], Wy2f, by2f, fY[1], ND, ED, MPY);
  gc(xpadY, di_e[1], di_sim[1], Wy1s, by1s, fY[2], ND, ED, MPY);
  gc(fY[2], di_e[1], di_sim[1], Wy2s, by2s, fY[3], ND, ED, MPY);
  gc(xpadY, di_e[2], di_sim[2], Wy1s, by1s, fY[4], ND, ED, MPY);
  gc(fY[2], di_e[2], di_sim[2], Wy2s, by2s, fY[5], ND, ED, MPY);

  // channel attention, x side; fold lin_w into zx so each pair is one dot
  mmgcn_pool_mean<<<6, 256, 0, stream>>>(featX, pooled, NFX, CSX);
  mmgcn_attn_mlp<<<1, 32, 0, stream>>>(pooled, fc1x_w, fc1x_b, fc2x_w, fc2x_b,
                                       svec);
  mmgcn_combine<<<cdiv(NFX, 256), 256, 0, stream>>>(featX, svec, cnnx_w,
                                                    cnnx_b, lin_w, zx, NFX,
                                                    CSX);
  // y side
  mmgcn_pool_mean<<<6, 256, 0, stream>>>(featY, pooled, NFY, CSY);
  mmgcn_attn_mlp<<<1, 32, 0, stream>>>(pooled, fc1y_w, fc1y_b, fc2y_w, fc2y_b,
                                       svec);
  mmgcn_combine<<<cdiv(NFY, 256), 256, 0, stream>>>(featY, svec, cnny_w,
                                                    cnny_b, nullptr, yfin, NFY,
                                                    CSY);

  // 1M-pair scoring, feature tables LDS-resident
  mmgcn_pair_score<<<1024, 256, SMEM_FLOATS * sizeof(float), stream>>>(
      train, zx, yfin, lin_b, (float*)d_out, P);
}